// LatentNet_64141041598717
// MI455X (gfx1250) — compile-verified
//
#include <hip/hip_runtime.h>
#include <math.h>

// ---------------------------------------------------------------------------
// LatentNet on MI455X (gfx1250): fused WMMA pipeline, f32 16x16x4 WMMA,
// with double-buffered GLOBAL_LOAD_ASYNC_TO_LDS_B128 staging (ASYNCcnt).
//   K0: xp = pad(x)  [8192 x 352]
//   K1: Wc = W2@W1 [64 x 342 pad 352], bc = W2@b1+b2
//   K2: z = x@Wc^T + bc  [8192 x 64]           (WMMA f32)
//   K3: sq[i] = ||z_i||^2
//   K4: fused  A-tile = sigmoid(-temp*dist+theta) (diag=1), rowsum,
//              h = rownorm(A) @ x               (WMMA f32, A never in HBM)
//   K5: logits = (h@W3^T+b3)@W6^T+b6 -> log_softmax
// ---------------------------------------------------------------------------

typedef float v2f __attribute__((ext_vector_type(2)));
typedef float v8f __attribute__((ext_vector_type(8)));

static __device__ __forceinline__ v8f wmma_f32(v2f a, v2f b, v8f c) {
  // (neg_a, A, neg_b, B, c_mod, C, reuse_a, reuse_b)
  return __builtin_amdgcn_wmma_f32_16x16x4_f32(false, a, false, b, (short)0, c,
                                               false, false);
}

// Async DMA: global -> LDS, 16 bytes per active lane, tracked by ASYNCcnt.
// GVS addressing: 64-bit SGPR base + 32-bit VGPR byte offset.
static __device__ __forceinline__ void async_ld_b128(unsigned lds_byte,
                                                     unsigned gl_off,
                                                     const void* base) {
  asm volatile("global_load_async_to_lds_b128 %0, %1, %2"
               :
               : "v"(lds_byte), "v"(gl_off), "s"(base)
               : "memory");
}

#define NROW 8192
#define DIN  342
#define DP   352    // padded feature dim (22 tiles of 16)
#define HD   64     // latent dim
#define XT_STRIDE 360  // LDS x-tile stride: 16B aligned, halves 16 banks apart
#define ZT_STRIDE 68   // LDS z-tile stride: 16B aligned, conflict-free b64 reads
#define AT_STRIDE 18   // LDS A-tile stride (even, distinct banks mod 64)

// ---------------- K0: pad x ----------------
__global__ void k_pad_x(const float* __restrict__ x, float* __restrict__ xp) {
  int idx = blockIdx.x * 256 + threadIdx.x;
  if (idx >= NROW * DP) return;
  int row = idx / DP, col = idx - row * DP;
  xp[idx] = (col < DIN) ? x[row * DIN + col] : 0.0f;
}

// ---------------- K1: collapse W2@W1, W2@b1+b2 ----------------
__global__ void k_wc(const float* __restrict__ W1, const float* __restrict__ b1,
                     const float* __restrict__ W2, const float* __restrict__ b2,
                     float* __restrict__ Wc, float* __restrict__ bc) {
  for (int idx = threadIdx.x; idx < HD * DP; idx += 256) {
    int c = idx / DP, d = idx - c * DP;
    float s = 0.0f;
    if (d < DIN)
      for (int k = 0; k < 128; ++k) s += W2[c * 128 + k] * W1[k * DIN + d];
    Wc[idx] = s;
  }
  if (threadIdx.x < HD) {
    int c = threadIdx.x;
    float s = b2[c];
    for (int k = 0; k < 128; ++k) s += W2[c * 128 + k] * b1[k];
    bc[c] = s;
  }
}

// ---------------- K2: z = x @ Wc^T + bc  (WMMA) ----------------
__global__ __launch_bounds__(256) void k_z(const float* __restrict__ xp,
                                           const float* __restrict__ Wc,
                                           const float* __restrict__ bc,
                                           float* __restrict__ z) {
  const int lane = threadIdx.x & 31;
  const int wave = threadIdx.x >> 5;
  const int l15 = lane & 15;
  const int h = lane >> 4;
  const int r0 = (blockIdx.x * 8 + wave) * 16;

  v8f acc[4];
#pragma unroll
  for (int t = 0; t < 4; ++t)
#pragma unroll
    for (int i = 0; i < 8; ++i) acc[t][i] = 0.0f;

  const float* arow = xp + (size_t)(r0 + l15) * DP + 2 * h;
  for (int s = 0; s < DP / 4; ++s) {
    v2f a = *(const v2f*)(arow + 4 * s);
#pragma unroll
    for (int t = 0; t < 4; ++t) {
      v2f b = *(const v2f*)(Wc + (size_t)(t * 16 + l15) * DP + 4 * s + 2 * h);
      acc[t] = wmma_f32(a, b, acc[t]);
    }
  }
#pragma unroll
  for (int t = 0; t < 4; ++t) {
    float bias = bc[t * 16 + l15];
#pragma unroll
    for (int r = 0; r < 8; ++r) {
      int m = r + 8 * h;
      z[(size_t)(r0 + m) * HD + t * 16 + l15] = acc[t][r] + bias;
    }
  }
}

// ---------------- K3: sq[i] = ||z_i||^2 ----------------
__global__ __launch_bounds__(256) void k_sq(const float* __restrict__ z,
                                            float* __restrict__ sq) {
  int row = blockIdx.x * 8 + (threadIdx.x >> 5);
  int lane = threadIdx.x & 31;
  float v0 = z[(size_t)row * HD + lane];
  float v1 = z[(size_t)row * HD + lane + 32];
  float s = v0 * v0 + v1 * v1;
#pragma unroll
  for (int off = 16; off; off >>= 1) s += __shfl_xor(s, off, 32);
  if (lane == 0) sq[row] = s;
}

// ---------------- K4: fused adjacency + aggregation ----------------
// Block = 256 threads = 8 waves, 64 rows (4 row-tiles x 16).
// wave: rt = wave>>1 (row tile), fh = wave&1 (feature half: 11 of 22 tiles).
// Column tiles stream through double-buffered LDS via async DMA.
__global__ __launch_bounds__(256) void k_fused(
    const float* __restrict__ xp, const float* __restrict__ z,
    const float* __restrict__ sq, const float* __restrict__ d_temp,
    const float* __restrict__ d_theta, float* __restrict__ hbuf) {
  __shared__ float sZ[2][16 * ZT_STRIDE];   // z[j:j+16, 0:64]
  __shared__ float sX[2][16 * XT_STRIDE];   // x[j:j+16, 0:352] (+zero pad)
  __shared__ float sA[8 * 16 * AT_STRIDE];  // per-wave 16x16 A-tile
  __shared__ float sRS[64];                 // rowsums for the block

  const int tid = threadIdx.x;
  const int lane = tid & 31, wave = tid >> 5;
  const int l15 = lane & 15, h = lane >> 4;
  const int rt = wave >> 1, fh = wave & 1;
  const int rowBase = blockIdx.x * 64 + rt * 16;
  const float temp = d_temp[0], theta = d_theta[0];

  // ---- precomputed per-thread DMA addresses (j-independent parts) ----
  // sZ: one b128 per thread: row = wave*2 + (lane>>4), c4 = lane&15
  const int zrow = wave * 2 + (lane >> 4);
  const int zc4 = lane & 15;
  const unsigned zG = (unsigned)(zrow * (HD * 4) + zc4 * 16);
  unsigned zL[2];
#pragma unroll
  for (int b = 0; b < 2; ++b)
    zL[b] = (unsigned)(size_t)&sZ[b][0] + (unsigned)(zrow * ZT_STRIDE + zc4 * 4) * 4u;
  // sX: 6 b128 per thread (k=5 only lanes 0..15): i = wave*176 + k*32 + lane
  unsigned xG[6], xL[2][6];
#pragma unroll
  for (int k = 0; k < 6; ++k) {
    int i = wave * 176 + k * 32 + lane;
    int xrow = i / (DP / 4), xc4 = i - xrow * (DP / 4);
    xG[k] = (unsigned)(xrow * (DP * 4) + xc4 * 16);
#pragma unroll
    for (int b = 0; b < 2; ++b)
      xL[b][k] = (unsigned)(size_t)&sX[b][0] +
                 (unsigned)(xrow * XT_STRIDE + xc4 * 4) * 4u;
  }

  // ---- zero the sX pad region (cols 352..359) once; DMA never touches it ----
  {
    int b = tid >> 7, rem = tid & 127;
    sX[b][(rem >> 3) * XT_STRIDE + DP + (rem & 7)] = 0.0f;
  }

  // Persistent A-operand fragments of z[rowBase:rowBase+16, 0:64]
  v2f zr[16];
  {
    const float* p = z + (size_t)(rowBase + l15) * HD + 2 * h;
#pragma unroll
    for (int s = 0; s < 16; ++s) zr[s] = *(const v2f*)(p + 4 * s);
  }
  float sqm[8];
#pragma unroll
  for (int r = 0; r < 8; ++r) sqm[r] = sq[rowBase + r + 8 * h];

  v8f acc[11];
#pragma unroll
  for (int t = 0; t < 11; ++t)
#pragma unroll
    for (int i = 0; i < 8; ++i) acc[t][i] = 0.0f;
  float rs[8] = {0, 0, 0, 0, 0, 0, 0, 0};
  float* aw = sA + wave * (16 * AT_STRIDE);

  // ---- prologue: async-stage tile j=0 into buffer 0 (7 ops per wave) ----
  {
    async_ld_b128(zL[0], zG, (const void*)z);
#pragma unroll
    for (int k = 0; k < 5; ++k) async_ld_b128(xL[0][k], xG[k], (const void*)xp);
    if (lane < 16) async_ld_b128(xL[0][5], xG[5], (const void*)xp);
  }

  for (int j = 0; j < NROW; j += 16) {
    const int cur = (j >> 4) & 1, nxt = cur ^ 1;
    __syncthreads();  // all readers of buffer 'nxt' (previous tile) are done
    const int jn = j + 16;
    if (jn < NROW) {
      // issue next tile into 'nxt' while we compute 'cur'
      unsigned jzb = (unsigned)jn * (HD * 4);
      unsigned jxb = (unsigned)jn * (DP * 4);
      async_ld_b128(zL[nxt], zG + jzb, (const void*)z);
#pragma unroll
      for (int k = 0; k < 5; ++k)
        async_ld_b128(xL[nxt][k], xG[k] + jxb, (const void*)xp);
      if (lane < 16) async_ld_b128(xL[nxt][5], xG[5] + jxb, (const void*)xp);
      // 7 newly issued remain in flight; previous batch (buffer 'cur') retires
      asm volatile("s_wait_asynccnt 0x7" ::: "memory");
    } else {
      asm volatile("s_wait_asynccnt 0x0" ::: "memory");
    }
    __syncthreads();  // buffer 'cur' fully visible to all waves

    const float* zc = sZ[cur];
    const float* xc = sX[cur];

    // --- S = Zrow @ Zcol^T  (16x16, K=64 via 16 WMMA f32 16x16x4) ---
    v8f S;
#pragma unroll
    for (int i = 0; i < 8; ++i) S[i] = 0.0f;
#pragma unroll
    for (int s = 0; s < 16; ++s) {
      v2f b = *(const v2f*)(zc + l15 * ZT_STRIDE + 4 * s + 2 * h);
      S = wmma_f32(zr[s], b, S);
    }

    // --- A-tile: sigmoid(-temp*dist + theta), diag=1, rowsum ---
    float sqj = sq[j + l15];
#pragma unroll
    for (int r = 0; r < 8; ++r) {
      float d = sqm[r] + sqj - 2.0f * S[r];
      d = fmaxf(d, 0.0f);
      float a = 1.0f / (1.0f + expf(temp * d - theta));
      if (rowBase + r + 8 * h == j + l15) a = 1.0f;
      rs[r] += a;
      aw[(r + 8 * h) * AT_STRIDE + l15] = a;  // D-layout -> LDS (m,k)
    }
    __builtin_amdgcn_wave_barrier();  // keep ds stores before ds reads

    // --- reload A-tile in A-operand layout ---
    v2f afr[4];
#pragma unroll
    for (int s2 = 0; s2 < 4; ++s2)
      afr[s2] = *(const v2f*)(aw + l15 * AT_STRIDE + 4 * s2 + 2 * h);

    // --- acc += A-tile @ x[j:j+16, featureTiles] ---
#pragma unroll
    for (int t = 0; t < 11; ++t) {
      int n0 = (fh * 11 + t) * 16 + l15;
#pragma unroll
      for (int s2 = 0; s2 < 4; ++s2) {
        int k0 = 4 * s2 + 2 * h;
        v2f b;
        b.x = xc[k0 * XT_STRIDE + n0];
        b.y = xc[(k0 + 1) * XT_STRIDE + n0];
        acc[t] = wmma_f32(afr[s2], b, acc[t]);
      }
    }
  }

  // --- rowsum: reduce across the 16 lanes of each half (fh==0 waves) ---
  if (fh == 0) {
#pragma unroll
    for (int r = 0; r < 8; ++r) {
      float v = rs[r];
      v += __shfl_xor(v, 1, 16);
      v += __shfl_xor(v, 2, 16);
      v += __shfl_xor(v, 4, 16);
      v += __shfl_xor(v, 8, 16);
      if (l15 == 0) sRS[rt * 16 + r + 8 * h] = v;
    }
  }
  __syncthreads();

  // --- normalize rows and store h ---
#pragma unroll
  for (int r = 0; r < 8; ++r) {
    int m = r + 8 * h;
    float inv = 1.0f / sRS[rt * 16 + m];
    size_t row = (size_t)(rowBase + m) * DP;
#pragma unroll
    for (int t = 0; t < 11; ++t) {
      int n = (fh * 11 + t) * 16 + l15;
      hbuf[row + n] = acc[t][r] * inv;
    }
  }
}

// ---------------- K5: head  logits -> log_softmax ----------------
__global__ __launch_bounds__(256) void k_head(
    const float* __restrict__ hbuf, const float* __restrict__ W3,
    const float* __restrict__ b3, const float* __restrict__ W6,
    const float* __restrict__ b6, float* __restrict__ out) {
  int row = blockIdx.x * 8 + (threadIdx.x >> 5);
  int c = threadIdx.x & 31;
  const float* hr = hbuf + (size_t)row * DP;
  const float* w = W3 + (size_t)c * DIN;
  float accv = b3[c];
  for (int f = 0; f < DIN; ++f) accv = fmaf(hr[f], w[f], accv);
  float t0 = accv * W6[c];
  float t1 = accv * W6[32 + c];
#pragma unroll
  for (int off = 16; off; off >>= 1) {
    t0 += __shfl_xor(t0, off, 32);
    t1 += __shfl_xor(t1, off, 32);
  }
  float l0 = t0 + b6[0], l1 = t1 + b6[1];
  float m = fmaxf(l0, l1);
  float lse = m + logf(expf(l0 - m) + expf(l1 - m));
  if (c == 0) {
    out[row * 2 + 0] = l0 - lse;
    out[row * 2 + 1] = l1 - lse;
  }
}

// ---------------------------------------------------------------------------
extern "C" void kernel_launch(void* const* d_in, const int* in_sizes, int n_in,
                              void* d_out, int out_size, void* d_ws,
                              size_t ws_size, hipStream_t stream) {
  const float* x    = (const float*)d_in[0];
  const float* W1   = (const float*)d_in[1];
  const float* b1   = (const float*)d_in[2];
  const float* W2   = (const float*)d_in[3];
  const float* b2   = (const float*)d_in[4];
  const float* W3   = (const float*)d_in[5];
  const float* b3   = (const float*)d_in[6];
  const float* W6   = (const float*)d_in[7];
  const float* b6   = (const float*)d_in[8];
  const float* temp = (const float*)d_in[9];
  const float* theta= (const float*)d_in[10];
  float* out = (float*)d_out;

  float* ws = (float*)d_ws;
  float* Wc = ws;                        // 64*352
  float* bc = Wc + HD * DP;              // 64
  float* xp = bc + HD;                   // 8192*352
  float* zb = xp + (size_t)NROW * DP;    // 8192*64
  float* sq = zb + (size_t)NROW * HD;    // 8192
  float* hb = sq + NROW;                 // 8192*352
  // total ~25.3 MB of workspace

  k_pad_x<<<(NROW * DP + 255) / 256, 256, 0, stream>>>(x, xp);
  k_wc<<<1, 256, 0, stream>>>(W1, b1, W2, b2, Wc, bc);
  k_z<<<NROW / (16 * 8), 256, 0, stream>>>(xp, Wc, bc, zb);
  k_sq<<<NROW / 8, 256, 0, stream>>>(zb, sq);
  k_fused<<<NROW / 64, 256, 0, stream>>>(xp, zb, sq, temp, theta, hb);
  k_head<<<NROW / 8, 256, 0, stream>>>(hb, W3, b3, W6, b6, out);
}